// LSSCVA_68101001445703
// MI455X (gfx1250) — compile-verified
//
#include <hip/hip_runtime.h>

// ---- LSS constants ----
#define B_   4
#define N_   6
#define D_   41
#define FH_  16
#define FW_  44
#define C_   64
#define NX_  200
#define NY_  200
#define NP_  (B_*N_*D_*FH_*FW_)        // 692736 points
#define GRID_PER_B_ (NX_*NY_)          // 40000 voxels per batch (NZ=1)
#define OUT_PER_B_  (C_*GRID_PER_B_)   // 2560000 floats per batch
#define GRID_ELEMS_ (B_*GRID_PER_B_*C_) // 10,240,000 floats

typedef __attribute__((ext_vector_type(2))) float v2f;
typedef __attribute__((ext_vector_type(8))) float v8f;

// Guaranteed-native f32 atomic add (no CAS-loop fallback risk).
// Non-returning global atomic; tracked by STOREcnt, implicit wait at s_endpgm.
__device__ __forceinline__ void atomic_add_f32_native(float* p, float v) {
    asm volatile("global_atomic_add_f32 %0, %1, off"
                 :
                 : "v"(p), "v"(v)
                 : "memory");
}

// 3x3 inverse via adjugate
__device__ __forceinline__ void inv3(const float* m, float* o) {
    float a=m[0],b=m[1],c=m[2],d=m[3],e=m[4],f=m[5],g=m[6],h=m[7],i=m[8];
    float A =  e*i - f*h;
    float Bq = -(d*i - f*g);
    float Cq =  d*h - e*g;
    float det = a*A + b*Bq + c*Cq;
    float r = 1.0f/det;
    o[0]=A*r;              o[1]=-(b*i-c*h)*r;    o[2]=(b*f-c*e)*r;
    o[3]=Bq*r;             o[4]=(a*i-c*g)*r;     o[5]=-(a*f-c*d)*r;
    o[6]=Cq*r;             o[7]=-(a*h-b*g)*r;    o[8]=(a*e-b*d)*r;
}

// Kernel A: per-(b,n) params (24 entries x 24 floats):
//   [0..8]  R1 = inv(post_rots)      [9..11]  t1 = -R1 @ post_trans
//   [12..20] comb = rots @ inv(intrins)   [21..23] trans
__global__ void lss_params(const float* __restrict__ rots,
                           const float* __restrict__ trans,
                           const float* __restrict__ intrins,
                           const float* __restrict__ post_rots,
                           const float* __restrict__ post_trans,
                           float* __restrict__ params) {
    int t = threadIdx.x;
    if (t >= B_*N_) return;
    const float* R  = rots       + t*9;
    const float* Tr = trans      + t*3;
    const float* Kk = intrins    + t*9;
    const float* PR = post_rots  + t*9;
    const float* PT = post_trans + t*3;
    float Ki[9], R1[9];
    inv3(Kk, Ki);
    inv3(PR, R1);
    float* P = params + t*24;
    for (int k = 0; k < 9; ++k) P[k] = R1[k];
    for (int r = 0; r < 3; ++r)
        P[9+r] = -(R1[r*3+0]*PT[0] + R1[r*3+1]*PT[1] + R1[r*3+2]*PT[2]);
    for (int r = 0; r < 3; ++r)
        for (int c = 0; c < 3; ++c)
            P[12 + r*3 + c] = R[r*3+0]*Ki[0*3+c] + R[r*3+1]*Ki[1*3+c] + R[r*3+2]*Ki[2*3+c];
    for (int r = 0; r < 3; ++r) P[21+r] = Tr[r];
}

// Kernel B: geometry via WMMA. Each wave32 handles 16 points:
//   stage1 (scalar, branchless): p = R1@f + t1, then q = (px*pz, py*pz, pz, 1)
//   stage2 (WMMA 16x16x4 f32):  D = A(16x4 affine [comb|trans]) x B(4x16 points q)
// D rows 0..2 (lanes 0..15) = geom xyz of point N=lane.
__global__ __launch_bounds__(256) void lss_geom_wmma(const float* __restrict__ params,
                                                     int* __restrict__ idx) {
    int tid  = blockIdx.x * blockDim.x + threadIdx.x;
    int lane = threadIdx.x & 31;
    int wave = tid >> 5;
    int m    = lane & 15;
    int p    = wave * 16 + m;              // always < NP_ (exact launch)

    // decode point -> (bn, d, h, w); all 16 points of a tile share bn
    int w  = p % FW_;  int t = p / FW_;
    int h  = t % FH_;  t /= FH_;
    int dd = t % D_;   t /= D_;            // t = b*N_ + n
    const float* P = params + t * 24;

    float u   = (float)w * (351.0f / 43.0f);
    float v   = (float)h * (127.0f / 15.0f);
    float dep = 4.0f + (float)dd;

    // stage 1: post-transform (all 32 lanes compute for point lane&15)
    float px = P[0]*u + P[1]*v + P[2]*dep + P[9];
    float py = P[3]*u + P[4]*v + P[5]*dep + P[10];
    float pz = P[6]*u + P[7]*v + P[8]*dep + P[11];

    bool lo = lane < 16;

    // A matrix (16x4): lane L<16 holds A[M=L][K=0],A[M=L][K=1];
    // lane L>=16 holds A[M=L-16][K=2],A[M=L-16][K=3]. Rows >=3 are zero.
    int  r3   = (m < 3) ? m : 0;
    float msk = (m < 3) ? 1.0f : 0.0f;
    float c0 = P[12 + r3*3 + 0];
    float c1 = P[12 + r3*3 + 1];
    float c2 = P[12 + r3*3 + 2];
    float tr = P[21 + r3];
    v2f A;
    A.x = (lo ? c0 : c2) * msk;
    A.y = (lo ? c1 : tr) * msk;

    // B matrix (4x16): lane L<16 -> B[0][L]=px*pz, B[1][L]=py*pz;
    // lane L>=16 -> B[2][L-16]=pz, B[3][L-16]=1
    v2f Bv;
    Bv.x = lo ? px * pz : pz;
    Bv.y = lo ? py * pz : 1.0f;

    v8f C = {};
    C = __builtin_amdgcn_wmma_f32_16x16x4_f32(false, A, false, Bv,
                                              (short)0, C, false, false);

    if (lo) {
        // trunc-toward-zero matches .astype(int32)
        float vx = (C[0] + 50.0f) * 2.0f;
        float vy = (C[1] + 50.0f) * 2.0f;
        float vz = (C[2] + 10.0f) * 0.05f;
        int gx = (int)vx, gy = (int)vy, gz = (int)vz;
        bool kept = (gx >= 0) && (gx < NX_) && (gy >= 0) && (gy < NY_) && (gz == 0);
        int b = t / N_;
        idx[p] = kept ? (b * GRID_PER_B_ + gx * NY_ + gy) : -1;
    }
}

// b128 zero fill
__global__ void lss_zero4(float4* __restrict__ p, int n4) {
    int i = blockIdx.x * blockDim.x + threadIdx.x;
    if (i < n4) p[i] = make_float4(0.f, 0.f, 0.f, 0.f);
}

// Kernel C: scatter-add. One thread per (point, channel).
// channelLast=1: dst[flat*64 + c]  (contiguous 256B per point -> coalesced L2 atomics)
// channelLast=0: dst[b*OUT_PER_B + c*GRID_PER_B + (flat % GRID_PER_B)] (final layout)
__global__ __launch_bounds__(256) void lss_scatter(const float* __restrict__ x,
                                                   const int* __restrict__ idx,
                                                   float* __restrict__ dst,
                                                   int channelLast) {
    int tid = blockIdx.x * blockDim.x + threadIdx.x;   // < NP_*C_ (exact)
    int c   = tid & 63;
    int p   = tid >> 6;
    int fl  = idx[p];
    if (fl < 0) return;
    float val = x[tid];
    int off;
    if (channelLast) {
        off = fl * 64 + c;
    } else {
        int b   = fl / GRID_PER_B_;
        int rem = fl - b * GRID_PER_B_;
        off = b * OUT_PER_B_ + c * GRID_PER_B_ + rem;
    }
    atomic_add_f32_native(dst + off, val);
}

// Kernel D: [s=(b,x,y)][c] -> [b][c][x,y] transpose, LDS-tiled (32 s x 64 c per block)
__global__ __launch_bounds__(256) void lss_transpose(const float* __restrict__ scratch,
                                                     float* __restrict__ out) {
    __shared__ float tile[32 * 65];
    int blk   = blockIdx.x;
    int b     = blk / (GRID_PER_B_ / 32);
    int srow0 = (blk % (GRID_PER_B_ / 32)) * 32;
    const float* src = scratch + ((size_t)b * GRID_PER_B_ + srow0) * 64;
    int t = threadIdx.x;
    int c = t & 63, i = t >> 6;
    #pragma unroll
    for (int j = 0; j < 8; ++j) {
        int r = i * 8 + j;
        tile[r * 65 + c] = src[r * 64 + c];
    }
    __syncthreads();
    int so = t & 31, cc = t >> 5;
    float* dstB = out + (size_t)b * OUT_PER_B_;
    #pragma unroll
    for (int j = 0; j < 8; ++j) {
        int ch = cc * 8 + j;
        dstB[ch * GRID_PER_B_ + srow0 + so] = tile[so * 65 + ch];
    }
}

extern "C" void kernel_launch(void* const* d_in, const int* in_sizes, int n_in,
                              void* d_out, int out_size, void* d_ws, size_t ws_size,
                              hipStream_t stream) {
    const float* x          = (const float*)d_in[0];
    const float* rots       = (const float*)d_in[1];
    const float* trans      = (const float*)d_in[2];
    const float* intrins    = (const float*)d_in[3];
    const float* post_rots  = (const float*)d_in[4];
    const float* post_trans = (const float*)d_in[5];
    float* out = (float*)d_out;

    char*  ws     = (char*)d_ws;
    float* params = (float*)ws;                                  // 576 floats
    int*   idx    = (int*)(ws + 4096);                           // NP_ ints
    size_t scratchOff = 4096 + (((size_t)NP_ * 4 + 255) / 256) * 256;
    float* scratch = (float*)(ws + scratchOff);
    int channelLast = (ws_size >= scratchOff + (size_t)GRID_ELEMS_ * 4) ? 1 : 0;

    lss_params<<<1, 32, 0, stream>>>(rots, trans, intrins, post_rots, post_trans, params);

    // 692736 points / 16 per wave = 43296 waves -> 1,385,472 threads -> 5412 blocks
    lss_geom_wmma<<<(NP_ / 16 * 32) / 256, 256, 0, stream>>>(params, idx);

    int n4 = GRID_ELEMS_ / 4;   // 2,560,000 float4
    if (channelLast) {
        lss_zero4<<<(n4 + 255) / 256, 256, 0, stream>>>((float4*)scratch, n4);
        lss_scatter<<<(NP_ * C_) / 256, 256, 0, stream>>>(x, idx, scratch, 1);
        lss_transpose<<<B_ * (GRID_PER_B_ / 32), 256, 0, stream>>>(scratch, out);
    } else {
        lss_zero4<<<(n4 + 255) / 256, 256, 0, stream>>>((float4*)out, n4);
        lss_scatter<<<(NP_ * C_) / 256, 256, 0, stream>>>(x, idx, out, 0);
    }
}